// BilateralFilterModule_56495999811800
// MI455X (gfx1250) — compile-verified
//
#include <hip/hip_runtime.h>
#include <stdint.h>

// Bilateral filter, K=7, sigma_color=0.1, image 720x1280x3 fp32 (NCHW planar).
// Strategy: stage haloed tile into LDS via gfx1250 async global->LDS loads
// (ASYNCcnt path), zero-fill halo OOB (reference zero-pads), 49-tap unrolled
// compute from LDS. Spatial gaussian g is tiled-constant per plane -> load 49
// scalars, not the 180MB array. norm_color cancels in the normalization.

#define KW 7
#define PAD 3
#define H_IMG 720
#define W_IMG 1280
#define BX 32
#define BY 8
#define TW (BX + 2 * PAD) /* 38 */
#define THL (BY + 2 * PAD) /* 14 */

__device__ __forceinline__ void async_load_f32(const float* gsrc, float* ldst) {
#if defined(__has_builtin) && __has_builtin(__builtin_amdgcn_global_load_async_to_lds_b32)
  typedef __attribute__((address_space(1))) int* gptr_t;
  typedef __attribute__((address_space(3))) int* lptr_t;
  __builtin_amdgcn_global_load_async_to_lds_b32(
      (gptr_t)(uintptr_t)gsrc, (lptr_t)(uint32_t)(uintptr_t)ldst, 0, 0);
#else
  uint32_t lds_off = (uint32_t)(uintptr_t)ldst; // low 32 bits of flat LDS ptr = LDS byte addr
  asm volatile("global_load_async_to_lds_b32 %0, %1, off"
               :
               : "v"(lds_off), "v"(gsrc)
               : "memory");
#endif
}

__device__ __forceinline__ void wait_asynccnt0() {
#if defined(__has_builtin) && __has_builtin(__builtin_amdgcn_s_wait_asynccnt)
  __builtin_amdgcn_s_wait_asynccnt(0);
#else
  asm volatile("s_wait_asynccnt 0x0" ::: "memory");
#endif
}

__global__ __launch_bounds__(BX* BY) void bilateral_7x7_kernel(
    const float* __restrict__ I, const float* __restrict__ g,
    float* __restrict__ out) {
  __shared__ float tile[3][THL][TW];
  __shared__ float gk[KW * KW];

  const int tx = threadIdx.x; // 0..31
  const int ty = threadIdx.y; // 0..7
  const int tid = ty * BX + tx;
  const int x0 = (int)blockIdx.x * BX - PAD;
  const int y0 = (int)blockIdx.y * BY - PAD;
  const size_t plane = (size_t)H_IMG * W_IMG;

  // Spatial gaussian: each of the 49 planes of g is a replicated constant;
  // read element 0 of each plane.
  if (tid < KW * KW) gk[tid] = g[(size_t)tid * plane];

  // Stage 3-channel haloed tile. In-bounds -> async global->LDS; OOB -> 0
  // (reference zero-pads, and zero taps still contribute to the denominator).
  const int NT = 3 * THL * TW; // 1596
  for (int i = tid; i < NT; i += BX * BY) {
    const int c = i / (THL * TW);
    const int rem = i - c * (THL * TW);
    const int r = rem / TW;
    const int col = rem - r * TW;
    const int gy = y0 + r;
    const int gx = x0 + col;
    float* lp = &tile[c][r][col];
    if ((unsigned)gy < (unsigned)H_IMG && (unsigned)gx < (unsigned)W_IMG) {
      async_load_f32(I + (size_t)c * plane + (size_t)gy * W_IMG + gx, lp);
    } else {
      *lp = 0.0f;
    }
  }
  wait_asynccnt0(); // wave's async tile data landed in LDS
  __syncthreads();  // all waves' data + zero-fills visible

  const float cR = tile[0][ty + PAD][tx + PAD];
  const float cG = tile[1][ty + PAD][tx + PAD];
  const float cB = tile[2][ty + PAD][tx + PAD];

  const float negInv2Sig = -50.0f; // -1/(2*0.1^2)
  float wsum = 0.0f, aR = 0.0f, aG = 0.0f, aB = 0.0f;
#pragma unroll
  for (int dy = 0; dy < KW; ++dy) {
#pragma unroll
    for (int dx = 0; dx < KW; ++dx) {
      const float r = tile[0][ty + dy][tx + dx];
      const float gc = tile[1][ty + dy][tx + dx];
      const float b = tile[2][ty + dy][tx + dx];
      const float dr = r - cR;
      const float dg = gc - cG;
      const float db = b - cB;
      const float D = fmaf(dr, dr, fmaf(dg, dg, db * db));
      const float w = __expf(D * negInv2Sig) * gk[dy * KW + dx];
      wsum += w;
      aR = fmaf(w, r, aR);
      aG = fmaf(w, gc, aG);
      aB = fmaf(w, b, aB);
    }
  }

  const int gx = (int)blockIdx.x * BX + tx;
  const int gy = (int)blockIdx.y * BY + ty;
  const float inv = 1.0f / wsum;
  const size_t p = (size_t)gy * W_IMG + gx;
  out[p] = aR * inv;
  out[plane + p] = aG * inv;
  out[2 * plane + p] = aB * inv;
}

extern "C" void kernel_launch(void* const* d_in, const int* in_sizes, int n_in,
                              void* d_out, int out_size, void* d_ws,
                              size_t ws_size, hipStream_t stream) {
  (void)in_sizes;
  (void)n_in;
  (void)out_size;
  (void)d_ws;
  (void)ws_size;
  const float* I = (const float*)d_in[0];
  const float* g = (const float*)d_in[1];
  float* out = (float*)d_out;

  dim3 block(BX, BY);                      // 256 threads = 8 wave32
  dim3 grid(W_IMG / BX, H_IMG / BY);       // 40 x 90, exact cover
  bilateral_7x7_kernel<<<grid, block, 0, stream>>>(I, g, out);
}